// PositionalEncoding_63479616634958
// MI455X (gfx1250) — compile-verified
//
#include <hip/hip_runtime.h>
#include <hip/hip_bf16.h>

// Problem constants (from the reference):
//   S=4096 seq, B=16 batch, D=256 d_model, N_CHAINS=8, MAX_LEN=5000
// out[s,b,:] = x[s,b,:] + pe[pos(s,b), :]
// chain_index is sorted along seq per column -> pos = s - lower_bound(chain, column b)

namespace {
constexpr int S        = 4096;
constexpr int B        = 16;
constexpr int D        = 256;
constexpr int NCHAINS  = 8;
constexpr int MAXLEN   = 5000;
constexpr int D4       = D / 4;        // 64 float4 per row
constexpr int ROWS     = S * B;        // 65536 rows
constexpr int THREADS  = 256;          // 8 waves (wave32) per block
constexpr int ROWS_PER_BLOCK = THREADS / D4;  // 4 rows per block per iteration
constexpr int BLOCKS   = 4096;         // grid-stride: 65536 / (4096*4) = 4 iterations
}

typedef __attribute__((ext_vector_type(4))) float v4f;

// ---------------------------------------------------------------------------
// Kernel 1: per (chain c, batch b) compute start[c*B+b] = #elements < c in the
// sorted column b of chain_index, via binary search (lower bound). 128 threads.
// ---------------------------------------------------------------------------
__global__ void pe_chain_start_kernel(const int* __restrict__ chain,
                                      int* __restrict__ start) {
    const int t = threadIdx.x;
    if (t >= NCHAINS * B) return;
    const int c = t / B;
    const int b = t - c * B;
    int lo = 0, hi = S;
    // lower bound of value c in chain[:, b] (stride B, sorted non-decreasing)
    while (lo < hi) {
        const int mid = (lo + hi) >> 1;
        const int v = chain[mid * B + b];
        if (v < c) lo = mid + 1; else hi = mid;
    }
    start[c * B + b] = lo;
}

// ---------------------------------------------------------------------------
// Kernel 2: streaming gather-add.
//   Each block iteration covers ROWS_PER_BLOCK rows; 64 lanes per row each own
//   one float4 (16 B) -> fully coalesced B128 accesses.
//   x: non-temporal load (read once), out: non-temporal store (write once),
//   pe: regular load so the 5 MB table stays hot in L2 (192 MB).
//   Next iteration's x tile is prefetched (global_prefetch_b8, gfx1250 path).
// ---------------------------------------------------------------------------
__global__ void pe_add_kernel(const v4f* __restrict__ x4,
                              const v4f* __restrict__ pe4,
                              const int* __restrict__ chain,
                              const int* __restrict__ start,
                              v4f* __restrict__ out4) {
    const int d4       = threadIdx.x & (D4 - 1);   // 0..63, float4 index in row
    const int rowInBlk = threadIdx.x >> 6;         // 0..3
    const int rowStride = BLOCKS * ROWS_PER_BLOCK; // rows advanced per iteration

    int row = blockIdx.x * ROWS_PER_BLOCK + rowInBlk;

    #pragma unroll 1
    for (; row < ROWS; row += rowStride) {
        // Prefetch next iteration's x chunk into the cache hierarchy.
        {
            const int nrow = row + rowStride;
            if (nrow < ROWS) {
                const long nidx = (long)nrow * D4 + d4;
                __builtin_prefetch((const void*)(x4 + nidx), 0, 0);
            }
        }

        const int s = row >> 4;          // row / B
        const int b = row & (B - 1);     // row % B

        const int c = chain[row];        // broadcast across the 64 row lanes
        int pos;
        if (c < 0) {
            pos = 0;                     // masked rows in the torch module
        } else {
            pos = s - start[c * B + b];
            pos = (pos < 0) ? 0 : pos;
        }
        pos = (pos > MAXLEN - 1) ? (MAXLEN - 1) : pos;

        const long i = (long)row * D4 + d4;

        const v4f xv = __builtin_nontemporal_load(x4 + i);       // streamed, NT
        const v4f pv = pe4[(long)pos * D4 + d4];                 // L2-resident

        v4f o;
        o.x = xv.x + pv.x;
        o.y = xv.y + pv.y;
        o.z = xv.z + pv.z;
        o.w = xv.w + pv.w;

        __builtin_nontemporal_store(o, out4 + i);                // streamed, NT
    }
}

// ---------------------------------------------------------------------------
// Launcher. d_in order per setup_inputs(): x (f32), pe (f32), chain_index (i32)
// d_out: f32 [S,B,D]. d_ws: 512 B used for the start table.
// ---------------------------------------------------------------------------
extern "C" void kernel_launch(void* const* d_in, const int* in_sizes, int n_in,
                              void* d_out, int out_size, void* d_ws, size_t ws_size,
                              hipStream_t stream) {
    (void)in_sizes; (void)n_in; (void)out_size; (void)ws_size;

    const float* x      = (const float*)d_in[0];
    const float* pe     = (const float*)d_in[1];
    const int*   chain  = (const int*)d_in[2];
    float*       out    = (float*)d_out;
    int*         start  = (int*)d_ws;   // NCHAINS*B = 128 ints

    pe_chain_start_kernel<<<1, 128, 0, stream>>>(chain, start);

    pe_add_kernel<<<BLOCKS, THREADS, 0, stream>>>(
        (const v4f*)x, (const v4f*)pe, chain, start, (v4f*)out);
}